// sdp_38156489458322
// MI455X (gfx1250) — compile-verified
//
#include <hip/hip_runtime.h>

typedef __attribute__((ext_vector_type(2))) float v2f;
typedef __attribute__((ext_vector_type(4))) float v4f;
typedef __attribute__((ext_vector_type(8))) float v8f;

#define B_   4
#define H_   12
#define T_   2048
#define D_   64
#define NT   (T_ / 16)          // 128 tiles of 16 along seq
#define SCALE 0.125f            // 1/sqrt(64), folded into A fragments
#define MASKVAL (-1.0e30f)

// One 16x16 fp32 score tile S = Qtile * Ktile^T via 16 chained
// V_WMMA_F32_16X16X4_F32 (K accumulated 4 at a time over D=64).
// A layout (16x4 f32): element (M,K) at lane = M + 16*(K>>1), vgpr = K&1.
// B layout (4x16 f32): element (K,N) at lane = N + 16*(K>>1), vgpr = K&1,
// which for B = Ktile^T means identical row-major addressing on K.
__device__ __forceinline__ v8f score_tile(const v2f a[16],
                                          const float* __restrict__ kbase,
                                          int lane) {
  const int n  = lane & 15;
  const int dh = (lane >> 4) << 1;
  v8f c = {};
#pragma unroll
  for (int kk = 0; kk < 16; ++kk) {
    v2f b = *(const v2f*)(kbase + n * D_ + 4 * kk + dh);
    c = __builtin_amdgcn_wmma_f32_16x16x4_f32(
        false, a[kk], false, b, (short)0, c, false, false);
  }
  return c;
}

__global__ void __launch_bounds__(32)
causal_softmax_scores(const float* __restrict__ Q,
                      const float* __restrict__ K,
                      float* __restrict__ O) {
  const int blk  = blockIdx.x;
  const int jq   = blk % NT;        // q-tile index
  const int bh   = blk / NT;        // fused (b,h)
  const int lane = threadIdx.x;

  const float* qbase = Q + ((size_t)bh * T_ + (size_t)jq * 16) * D_;
  const float* kb0   = K + (size_t)bh * T_ * D_;
  float*       obase = O + ((size_t)bh * T_ + (size_t)jq * 16) * (size_t)T_;

  // Preload all 16 A-fragments once; fold the 1/sqrt(D) scale in here.
  const int m  = lane & 15;
  const int dh = (lane >> 4) << 1;
  v2f a[16];
#pragma unroll
  for (int kk = 0; kk < 16; ++kk) {
    v2f v = *(const v2f*)(qbase + m * D_ + 4 * kk + dh);
    v.x *= SCALE; v.y *= SCALE;
    a[kk] = v;
  }

  const int h8 = (lane >> 4) << 3;  // high half holds rows M+8
  const int nc = lane & 15;         // this lane's column within a tile

  // Per-r causal mask flag for the diagonal tile (col > row within tile).
  bool dm[8];
#pragma unroll
  for (int r = 0; r < 8; ++r) dm[r] = nc > (r + h8);

  // ---------- Pass 1: row max (branchless, no exp) ----------
  float rm[8];
#pragma unroll
  for (int r = 0; r < 8; ++r) rm[r] = MASKVAL;

  for (int jk = 0; jk <= jq; ++jk) {
    v8f c = score_tile(a, kb0 + (size_t)jk * 16 * D_, lane);
    const bool diag = (jk == jq);
#pragma unroll
    for (int r = 0; r < 8; ++r) {
      float s = (diag && dm[r]) ? MASKVAL : c[r];
      rm[r] = fmaxf(rm[r], s);
    }
  }
#pragma unroll
  for (int off = 1; off <= 8; off <<= 1)
#pragma unroll
    for (int r = 0; r < 8; ++r)
      rm[r] = fmaxf(rm[r], __shfl_xor(rm[r], off, 32));

  // ---------- Pass 2: row sum of exp(s - max) ----------
  float rl[8];
#pragma unroll
  for (int r = 0; r < 8; ++r) rl[r] = 0.0f;

  for (int jk = 0; jk <= jq; ++jk) {
    v8f c = score_tile(a, kb0 + (size_t)jk * 16 * D_, lane);
    const bool diag = (jk == jq);
#pragma unroll
    for (int r = 0; r < 8; ++r) {
      float s = (diag && dm[r]) ? MASKVAL : c[r];
      rl[r] += __expf(s - rm[r]);   // underflows to 0 for masked entries
    }
  }
#pragma unroll
  for (int off = 1; off <= 8; off <<= 1)
#pragma unroll
    for (int r = 0; r < 8; ++r)
      rl[r] += __shfl_xor(rl[r], off, 32);

  float rinv[8];
#pragma unroll
  for (int r = 0; r < 8; ++r) rinv[r] = 1.0f / rl[r];

  // ---------- Pass 3: recompute, normalize, stream out ----------
  for (int jk = 0; jk <= jq; ++jk) {
    v8f c = score_tile(a, kb0 + (size_t)jk * 16 * D_, lane);
    const bool diag = (jk == jq);
#pragma unroll
    for (int r = 0; r < 8; ++r) {
      float s = (diag && dm[r]) ? MASKVAL : c[r];
      float p = __expf(s - rm[r]) * rinv[r];
      size_t off = (size_t)(r + h8) * T_ + (size_t)jk * 16 + nc;
      __builtin_nontemporal_store(p, obase + off);
    }
  }

  // ---------- Zero-fill the fully-masked upper-triangle columns ----------
  const int cstart = (jq + 1) * 16;     // 16-float aligned -> b128 aligned
  v4f z = {};
  for (int row = 0; row < 16; ++row) {
    float* rp = obase + (size_t)row * T_;
    for (int col = cstart + lane * 4; col < T_; col += 32 * 4)
      __builtin_nontemporal_store(z, (v4f*)(rp + col));
  }
}

extern "C" void kernel_launch(void* const* d_in, const int* in_sizes, int n_in,
                              void* d_out, int out_size, void* d_ws, size_t ws_size,
                              hipStream_t stream) {
  (void)in_sizes; (void)n_in; (void)out_size; (void)d_ws; (void)ws_size;
  const float* Q = (const float*)d_in[0];
  const float* K = (const float*)d_in[1];
  float*       O = (float*)d_out;

  dim3 grid(B_ * H_ * NT);   // 6144 waves: one per 16-row Q block
  dim3 block(32);            // one wave32 per workgroup
  hipLaunchKernelGGL(causal_softmax_scores, grid, block, 0, stream, Q, K, O);
}